// LiftSplatShootGPN_88656714925260
// MI455X (gfx1250) — compile-verified
//
#include <hip/hip_runtime.h>
#include <hip/hip_bf16.h>

// ---- problem dims (LiftSplatShootGPN defaults) ----
#define BB 8
#define NN 6
#define DDEP 41
#define FH 8
#define FW 22
#define CC 64
#define PTS_PER_CAM (DDEP * FH * FW)          // 7216 (divisible by 16)
#define NPTS (BB * NN * PTS_PER_CAM)          // 346368
#define NX0 200
#define NX1 200
#define PLANE (NX0 * NX1)                     // 40000
#define OUT_PER_B (CC * PLANE)                // 2,560,000

typedef __attribute__((ext_vector_type(2))) float v2f;
typedef __attribute__((ext_vector_type(8))) float v8f;

// ------------------------------------------------------------------
// Kernel 1: zero the output grid (float4 vectorized)
// ------------------------------------------------------------------
__global__ void zero_f32(float* __restrict__ p, int n) {
    int i = blockIdx.x * blockDim.x + threadIdx.x;
    int i4 = i * 4;
    if (i4 >= n) return;
    if (i4 + 4 <= n) {
        *(float4*)(p + i4) = make_float4(0.f, 0.f, 0.f, 0.f);
    } else {
        for (int k = i4; k < n; ++k) p[k] = 0.f;
    }
}

// ------------------------------------------------------------------
// Kernel 2: per-camera affine transforms into workspace.
//   mats[bn*24 + i*4 + {0,1,2}] = inv(post_rots)[i][:]   (stage 1)
//   mats[bn*24 + i*4 + 3]       = -(inv(post_rots) @ post_trans)[i]
//   mats[bn*24 + 12 + i*4 + {0,1,2}] = (rots @ inv(intrins))[i][:]
//   mats[bn*24 + 12 + i*4 + 3]       = trans[i]
// ------------------------------------------------------------------
__device__ __forceinline__ void inv3(const float* m, float* o) {
    float a = m[0], b = m[1], c = m[2];
    float d = m[3], e = m[4], f = m[5];
    float g = m[6], h = m[7], i = m[8];
    float det = a * (e * i - f * h) - b * (d * i - f * g) + c * (d * h - e * g);
    float r = 1.0f / det;
    o[0] = (e * i - f * h) * r;  o[1] = (c * h - b * i) * r;  o[2] = (b * f - c * e) * r;
    o[3] = (f * g - d * i) * r;  o[4] = (a * i - c * g) * r;  o[5] = (c * d - a * f) * r;
    o[6] = (d * h - e * g) * r;  o[7] = (b * g - a * h) * r;  o[8] = (a * e - b * d) * r;
}

__global__ void prep_mats(const float* __restrict__ rots,
                          const float* __restrict__ trans,
                          const float* __restrict__ intrins,
                          const float* __restrict__ post_rots,
                          const float* __restrict__ post_trans,
                          float* __restrict__ mats) {
    int bn = blockIdx.x * blockDim.x + threadIdx.x;
    if (bn >= BB * NN) return;
    float ipr[9], iin[9], cmb[9];
    inv3(post_rots + bn * 9, ipr);
    inv3(intrins + bn * 9, iin);
    const float* R = rots + bn * 9;
    for (int i = 0; i < 3; ++i)
        for (int j = 0; j < 3; ++j)
            cmb[i * 3 + j] = R[i * 3 + 0] * iin[0 * 3 + j] +
                             R[i * 3 + 1] * iin[1 * 3 + j] +
                             R[i * 3 + 2] * iin[2 * 3 + j];
    const float* pt = post_trans + bn * 3;
    const float* tr = trans + bn * 3;
    float* o = mats + bn * 24;
    for (int i = 0; i < 3; ++i) {
        o[i * 4 + 0] = ipr[i * 3 + 0];
        o[i * 4 + 1] = ipr[i * 3 + 1];
        o[i * 4 + 2] = ipr[i * 3 + 2];
        o[i * 4 + 3] = -(ipr[i * 3 + 0] * pt[0] + ipr[i * 3 + 1] * pt[1] + ipr[i * 3 + 2] * pt[2]);
        o[12 + i * 4 + 0] = cmb[i * 3 + 0];
        o[12 + i * 4 + 1] = cmb[i * 3 + 1];
        o[12 + i * 4 + 2] = cmb[i * 3 + 2];
        o[12 + i * 4 + 3] = tr[i];
    }
}

// ------------------------------------------------------------------
// Kernel 3: geometry via V_WMMA_F32_16X16X4_F32.
// Each wave32 handles 16 consecutive points of one camera.
//   A (16x4) = affine transform in rows 0..2 (rest zero):
//     lane m, vgpr0 = A[m][0] (lanes 0-15) / A[m][2] (lanes 16-31)
//     lane m, vgpr1 = A[m][1] (lanes 0-15) / A[m][3] (lanes 16-31)
//   B (4x16) = 16 homogeneous points:
//     vgpr0: lanes 0-15 = u_p, lanes 16-31 = depth_p
//     vgpr1: lanes 0-15 = v_p, lanes 16-31 = 1.0
//   D (16x16): lane p (<16), vgpr i = component i of point p  -> no shuffle
// Stage1 -> pointwise unproject -> Stage2 -> voxel index.
// ------------------------------------------------------------------
__global__ void geom_idx(const float* __restrict__ mats, int* __restrict__ flatout) {
    int t = blockIdx.x * blockDim.x + threadIdx.x;
    int wv = t >> 5;        // wave index; 16 points per wave
    int l = t & 31;         // lane
    int r = l & 15;         // point-in-wave / matrix row
    bool lo = (l < 16);

    int pg = wv * 16 + r;                       // global point id
    int bn = pg / PTS_PER_CAM;                  // camera id (constant per wave)
    int pi = pg - bn * PTS_PER_CAM;
    int d = pi / (FH * FW);
    int rem = pi - d * (FH * FW);
    int h = rem / FW;
    int w = rem - h * FW;
    float u   = (float)w * (351.0f / 21.0f);    // linspace(0, 351, 22)
    float v   = (float)h * (127.0f / 7.0f);     // linspace(0, 127, 8)
    float dep = 4.0f + (float)d;                // arange(4, 45)

    const float* A = mats + bn * 24;
    v2f a1, a2, b1, b2;
    a1.x = (r < 3) ? A[r * 4 + (lo ? 0 : 2)] : 0.0f;
    a1.y = (r < 3) ? A[r * 4 + (lo ? 1 : 3)] : 0.0f;
    a2.x = (r < 3) ? A[12 + r * 4 + (lo ? 0 : 2)] : 0.0f;
    a2.y = (r < 3) ? A[12 + r * 4 + (lo ? 1 : 3)] : 0.0f;
    b1.x = lo ? u : dep;
    b1.y = lo ? v : 1.0f;

    v8f c0 = {0.f, 0.f, 0.f, 0.f, 0.f, 0.f, 0.f, 0.f};
    // stage 1: p1 = inv(post_rots) @ (frustum - post_trans)
    v8f p1 = __builtin_amdgcn_wmma_f32_16x16x4_f32(false, a1, false, b1, (short)0, c0,
                                                   false, false);
    // pointwise unprojection; high lanes need z of their point from the low lane
    float zz = __shfl(p1[2], r, 32);
    b2.x = lo ? p1[0] * zz : zz;     // x*z | z
    b2.y = lo ? p1[1] * zz : 1.0f;   // y*z | 1
    // stage 2: geom = (rots @ inv(intrins)) @ q + trans
    v8f p2 = __builtin_amdgcn_wmma_f32_16x16x4_f32(false, a2, false, b2, (short)0, c0,
                                                   false, false);

    if (lo) {
        // (geom - (BX - DX/2)) / DX, truncated toward zero (matches astype(int32))
        int ix = (int)((p2[0] + 50.0f) / 0.5f);
        int iy = (int)((p2[1] + 50.0f) / 0.5f);
        int iz = (int)((p2[2] + 10.0f) / 20.0f);
        bool kept = (ix >= 0) & (ix < NX0) & (iy >= 0) & (iy < NX1) & (iz == 0);
        int b = pg / (NN * PTS_PER_CAM);
        // base output offset for channel 0 of this voxel in [B, C, 200, 200]
        flatout[pg] = kept ? (b * OUT_PER_B + ix * NX1 + iy) : -1;
    }
}

// ------------------------------------------------------------------
// Kernel 4: scatter-add features. 64 threads per point (one per channel).
// x_feats read is fully coalesced; atomic targets stride 40000 floats
// per channel but the whole 82MB output grid is L2-resident (192MB L2).
// ------------------------------------------------------------------
__global__ void scatter_add(const float* __restrict__ x,
                            const int* __restrict__ flat,
                            float* __restrict__ out) {
    int t = blockIdx.x * blockDim.x + threadIdx.x;
    int p = t >> 6;          // point id (4 points per 256-thread block)
    int c = t & 63;          // channel
    int f = flat[p];
    if (f < 0) return;
    float val = x[(size_t)p * CC + c];
    __hip_atomic_fetch_add(out + f + c * PLANE, val,
                           __ATOMIC_RELAXED, __HIP_MEMORY_SCOPE_AGENT);
}

// ------------------------------------------------------------------
extern "C" void kernel_launch(void* const* d_in, const int* in_sizes, int n_in,
                              void* d_out, int out_size, void* d_ws, size_t ws_size,
                              hipStream_t stream) {
    (void)in_sizes; (void)n_in;
    const float* x          = (const float*)d_in[0];
    const float* rots       = (const float*)d_in[1];
    const float* trans      = (const float*)d_in[2];
    const float* intrins    = (const float*)d_in[3];
    const float* post_rots  = (const float*)d_in[4];
    const float* post_trans = (const float*)d_in[5];
    float* out = (float*)d_out;

    // workspace layout: [0, 4608) camera affines, then per-point flat indices
    float* mats = (float*)d_ws;
    int*   flat = (int*)((char*)d_ws + (size_t)(BB * NN * 24 * sizeof(float)));
    (void)ws_size; // needs 4608 + 346368*4 ~= 1.39 MB

    // 1) zero output
    int zthreads = (out_size + 3) / 4;
    zero_f32<<<(zthreads + 255) / 256, 256, 0, stream>>>(out, out_size);

    // 2) per-camera transform prep (48 cameras)
    prep_mats<<<1, 64, 0, stream>>>(rots, trans, intrins, post_rots, post_trans, mats);

    // 3) WMMA geometry + voxel index: 346368/16 = 21648 waves, 8 waves/block
    geom_idx<<<(NPTS / 16) / 8, 256, 0, stream>>>(mats, flat);

    // 4) scatter-add: 4 points per block
    scatter_add<<<NPTS / 4, 256, 0, stream>>>(x, flat, out);
}